// HypergraphLearner_65000035058051
// MI455X (gfx1250) — compile-verified
//
#include <hip/hip_runtime.h>
#include <cstdint>
#include <cstddef>

// ---------------- problem dims ----------------
#define BB 8
#define NN 1152
#define DD 128
#define HH 8
#define LL 48
#define EE 24
#define KTILES (NN / 16)   // 72 key tiles
#define NEG_BIG (-1.0e9f)
#define NEG_INF (-3.0e38f)

typedef __attribute__((ext_vector_type(2))) float v2f;
typedef __attribute__((ext_vector_type(8))) float v8f;

__device__ __forceinline__ v8f wmma4(v2f a, v2f b, v8f c) {
  // D = A(16x4 f32) x B(4x16 f32) + C(16x16 f32)
  return __builtin_amdgcn_wmma_f32_16x16x4_f32(false, a, false, b, (short)0, c,
                                               false, false);
}

// =====================================================================
// Generic WMMA GEMM: out(M,128) = A(M,K) @ W(K,128) + bias, fused epilogue.
// MODE 0: out = acc + bias
// MODE 1: out = resid + relu(acc + bias)                  (Wo epilogue)
// MODE 2: out = relu(resid + acc + bias) * rowmask[m]     (quat epilogue)
// MODE 3: spike gate: tau=exp(ltau)+1e-4; decay=exp(-dt[m]/tau);
//         spike = ((acc+bias)*decay >= thr); out = resid*(1+spike)*rowmask[m]
// M must be a multiple of 16 (always true here). grid = (M/16, 8), block = 32.
// =====================================================================
template <int MODE>
__global__ __launch_bounds__(32) void gemm_k(
    const float* __restrict__ A, const float* __restrict__ W,
    const float* __restrict__ bias, float* __restrict__ out, int M, int K,
    const float* __restrict__ resid, const float* __restrict__ rowmask,
    const float* __restrict__ dtv, const float* __restrict__ ltau,
    const float* __restrict__ thr) {
  const int lane = threadIdx.x & 31;
  const int lh = lane >> 4;          // half-wave select
  const int lm = lane & 15;
  const int m0 = blockIdx.x << 4;
  const int n0 = blockIdx.y << 4;

  const float* arow = A + (size_t)(m0 + lm) * K + 2 * lh;
  v8f acc = {};
  for (int k0 = 0; k0 < K; k0 += 4) {
    v2f af;
    af.x = arow[k0];
    af.y = arow[k0 + 1];
    const float* wp = W + (size_t)(k0 + 2 * lh) * DD + n0 + lm;
    v2f bf;
    bf.x = wp[0];
    bf.y = wp[DD];
    acc = wmma4(af, bf, acc);
  }

#pragma unroll
  for (int r = 0; r < 8; ++r) {
    const int m = m0 + r + 8 * lh;
    const int n = n0 + lm;
    const size_t idx = (size_t)m * DD + n;
    float v = acc[r] + bias[n];
    if (MODE == 0) {
      out[idx] = v;
    } else if (MODE == 1) {
      out[idx] = resid[idx] + fmaxf(v, 0.0f);
    } else if (MODE == 2) {
      out[idx] = fmaxf(resid[idx] + v, 0.0f) * rowmask[m];
    } else {
      float tau = __expf(ltau[0]) + 1e-4f;
      float decay = __expf(-dtv[m] / tau);
      float sp = ((v * decay) >= thr[0]) ? 1.0f : 0.0f;
      out[idx] = resid[idx] * (1.0f + sp) * rowmask[m];
    }
  }
}

// =====================================================================
// Flash-style MHAB attention core.  One wave per (query-16-tile, head, batch).
//   O[q,:] = Qh[q,:] + softmax(mask(Qh Kh^T / sqrt(128))) @ Vh
// Q rows: b*S + q (S queries/batch); K,V rows: b*NN + n.  Head h = ch [16h,16h+16).
// MASKMODE 0: node     - valid iff xym_f[b,q]!=0 && xym_f[b,k]!=0   (mk = xym_f)
// MASKMODE 1: temporal - valid iff t_inc[b,q,k]!=0 && time_idx[b,k] <= q
// MASKMODE 2: variable - valid iff v_inc[b,q,k]!=0
// (layer-0 "soft" factor only rescales nonzero entries; mask tests ==0, so skip)
// =====================================================================
template <int MASKMODE>
__global__ __launch_bounds__(32) void attn_k(
    const float* __restrict__ Qb, const float* __restrict__ Kb,
    const float* __restrict__ Vb, float* __restrict__ Ob, int S,
    const float* __restrict__ mk, const int* __restrict__ tidx) {
  __shared__ float sc[16][17];  // tiny P-tile staging (C-layout -> A-layout)
  const int lane = threadIdx.x & 31;
  const int lh = lane >> 4;
  const int lm = lane & 15;
  const int b = blockIdx.z;
  const int h = blockIdx.y;
  const int q0 = blockIdx.x << 4;
  const int ch = h * 16;
  const size_t qbase = (size_t)b * S;
  const size_t kbase = (size_t)b * NN;
  const float scl = 0.088388347648318447f;  // 1/sqrt(128)

  // preload Q A-fragments (M=query, K=head-dim)
  int qr = q0 + lm;
  if (qr > S - 1) qr = S - 1;
  const float* qrow = Qb + (qbase + qr) * DD + ch;
  v2f aq[4];
#pragma unroll
  for (int j = 0; j < 4; ++j) {
    const int d = 4 * j + 2 * lh;
    aq[j].x = qrow[d];
    aq[j].y = qrow[d + 1];
  }

  float mx[8], sm[8];
  v8f of = {};
#pragma unroll
  for (int r = 0; r < 8; ++r) {
    mx[r] = NEG_INF;
    sm[r] = 0.0f;
  }

  for (int kt = 0; kt < KTILES; ++kt) {
    // ---- scores tile: 4 chained WMMAs over the 16-dim head ----
    v8f acc = {};
    const float* krow = Kb + (kbase + kt * 16 + lm) * DD + ch;
#pragma unroll
    for (int j = 0; j < 4; ++j) {
      const int d = 4 * j + 2 * lh;
      v2f bf;
      bf.x = krow[d];
      bf.y = krow[d + 1];
      acc = wmma4(aq[j], bf, acc);
    }
    __syncthreads();  // previous iteration's P reads done before overwrite
#pragma unroll
    for (int r = 0; r < 8; ++r) {
      const int m = r + 8 * lh;
      const int qg = q0 + m;
      const int kg = kt * 16 + lm;
      const int qgc = (qg > S - 1) ? (S - 1) : qg;
      float s = acc[r] * scl;
      bool valid = (qg < S);
      if (MASKMODE == 0) {
        valid = valid && (mk[(size_t)b * NN + qgc] != 0.0f) &&
                (mk[(size_t)b * NN + kg] != 0.0f);
      } else if (MASKMODE == 1) {
        valid = valid && (mk[((size_t)b * LL + qgc) * NN + kg] != 0.0f) &&
                (tidx[(size_t)b * NN + kg] <= qg);
      } else {
        valid = valid && (mk[((size_t)b * EE + qgc) * NN + kg] != 0.0f);
      }
      if (!valid) s = NEG_BIG;
      // online softmax, row stats reduced across the 16-lane half-wave
      float tmax = s;
#pragma unroll
      for (int o = 8; o > 0; o >>= 1) tmax = fmaxf(tmax, __shfl_xor(tmax, o, 16));
      const float mnew = fmaxf(mx[r], tmax);
      const float p = __expf(s - mnew);
      float ts = p;
#pragma unroll
      for (int o = 8; o > 0; o >>= 1) ts += __shfl_xor(ts, o, 16);
      const float corr = __expf(mx[r] - mnew);
      sm[r] = sm[r] * corr + ts;
      mx[r] = mnew;
      of[r] *= corr;
      sc[m][lm] = p;  // stage P tile
    }
    __syncthreads();
    // ---- O += P @ V : P re-read from LDS in A-layout ----
#pragma unroll
    for (int kk = 0; kk < 16; kk += 4) {
      const int key = kt * 16 + kk + 2 * lh;
      v2f af;
      af.x = sc[lm][kk + 2 * lh];
      af.y = sc[lm][kk + 2 * lh + 1];
      const float* vrow = Vb + (kbase + key) * DD + ch + lm;
      v2f bf;
      bf.x = vrow[0];
      bf.y = vrow[DD];
      of = wmma4(af, bf, of);
    }
  }

#pragma unroll
  for (int r = 0; r < 8; ++r) {
    const int m = r + 8 * lh;
    const int qg = q0 + m;
    if (qg < S) {
      const size_t idx = (qbase + qg) * DD + ch + lm;
      Ob[idx] = Qb[idx] + of[r] / sm[r];  // residual Qh + A@V
    }
  }
}

// =====================================================================
// Elementwise / gather helpers
// =====================================================================
__global__ void build_kin2_k(const float* __restrict__ he, int heS,
                             const int* __restrict__ idx,
                             const float* __restrict__ obs,
                             float* __restrict__ kin) {
  const size_t total = (size_t)BB * NN * 256;
  for (size_t t = (size_t)blockIdx.x * blockDim.x + threadIdx.x; t < total;
       t += (size_t)gridDim.x * blockDim.x) {
    const int c = (int)(t & 255);
    const size_t bn = t >> 8;
    const int b = (int)(bn / NN);
    float v;
    if (c < DD)
      v = he[((size_t)b * heS + idx[bn]) * DD + c];
    else
      v = obs[bn * DD + (c - DD)];
    kin[t] = v;
  }
}

__global__ void build_kin3_k(const float* __restrict__ the,
                             const int* __restrict__ tix,
                             const float* __restrict__ vhe,
                             const int* __restrict__ vix,
                             const float* __restrict__ obs,
                             float* __restrict__ kin) {
  const size_t total = (size_t)BB * NN * 384;
  for (size_t t = (size_t)blockIdx.x * blockDim.x + threadIdx.x; t < total;
       t += (size_t)gridDim.x * blockDim.x) {
    const int c = (int)(t % 384);
    const size_t bn = t / 384;
    const int b = (int)(bn / NN);
    float v;
    if (c < DD)
      v = the[((size_t)b * LL + tix[bn]) * DD + c];
    else if (c < 2 * DD)
      v = vhe[((size_t)b * EE + vix[bn]) * DD + (c - DD)];
    else
      v = obs[bn * DD + (c - 2 * DD)];
    kin[t] = v;
  }
}

// quaternion weight: wt[kd*128 + c] = W[c, kd], W = [[r,-i,-j,-k],[i,r,-k,j],
// [j,k,r,-i],[k,-j,i,r]] of (32 x 96) blocks; x @ W^T + b.
__global__ void build_quatw_k(const float* __restrict__ r,
                              const float* __restrict__ i_,
                              const float* __restrict__ j_,
                              const float* __restrict__ k_,
                              float* __restrict__ wt) {
  const int total = 384 * DD;
  for (int t = blockIdx.x * blockDim.x + threadIdx.x; t < total;
       t += gridDim.x * blockDim.x) {
    const int kd = t / DD;
    const int c = t % DD;
    const int br = c >> 5, cp = c & 31;
    const int bc = kd / 96, kp = kd % 96;
    const float* comp;
    float sgn;
    switch (br * 4 + bc) {
      case 0:  comp = r;  sgn = 1.f;  break;
      case 1:  comp = i_; sgn = -1.f; break;
      case 2:  comp = j_; sgn = -1.f; break;
      case 3:  comp = k_; sgn = -1.f; break;
      case 4:  comp = i_; sgn = 1.f;  break;
      case 5:  comp = r;  sgn = 1.f;  break;
      case 6:  comp = k_; sgn = -1.f; break;
      case 7:  comp = j_; sgn = 1.f;  break;
      case 8:  comp = j_; sgn = 1.f;  break;
      case 9:  comp = k_; sgn = 1.f;  break;
      case 10: comp = r;  sgn = 1.f;  break;
      case 11: comp = i_; sgn = -1.f; break;
      case 12: comp = k_; sgn = 1.f;  break;
      case 13: comp = j_; sgn = -1.f; break;
      case 14: comp = i_; sgn = 1.f;  break;
      default: comp = r;  sgn = 1.f;  break;
    }
    wt[t] = sgn * comp[cp * 96 + kp];
  }
}

// fine[b,e, l*2+d] = obs[b, l*E+e, d] * (xym[b,l,e] > 0)
__global__ void fine_k(const float* __restrict__ obs,
                       const float* __restrict__ xym,
                       float* __restrict__ fine) {
  const int total = BB * EE * 96;
  for (int t = blockIdx.x * blockDim.x + threadIdx.x; t < total;
       t += gridDim.x * blockDim.x) {
    const int x = t % 96;
    const int be = t / 96;
    const int e = be % EE;
    const int b = be / EE;
    const int l = x >> 1;
    const int d = x & 1;
    const int n = l * EE + e;
    const float g = (xym[((size_t)b * LL + l) * EE + e] > 0.0f) ? 1.0f : 0.0f;
    fine[t] = obs[((size_t)b * NN + n) * DD + d] * g;
  }
}

// _irr: tiny (24x24) merged-score attention, one block per batch
__global__ __launch_bounds__(32) void irr_k(
    const float* __restrict__ q, const float* __restrict__ k,
    const float* __restrict__ v, const float* __restrict__ fine,
    const float* __restrict__ thr, float* __restrict__ vhe) {
  const int b = blockIdx.x;
  const int t = threadIdx.x;
  if (t >= EE) return;
  float s[EE];
  const float th = thr[0];
  const float* qp = q + ((size_t)b * EE + t) * DD;
  const float* fq = fine + ((size_t)b * EE + t) * 96;
  for (int kk = 0; kk < EE; ++kk) {
    const float* kp = k + ((size_t)b * EE + kk) * DD;
    float dq = 0.0f;
    for (int c = 0; c < DD; ++c) dq += qp[c] * kp[c];
    dq *= 0.088388347648318447f;  // /sqrt(128)
    const float* fk = fine + ((size_t)b * EE + kk) * 96;
    float ax = 0.0f;
    for (int c = 0; c < 96; ++c) ax += fq[c] * fk[c];
    ax *= 0.10206207261596577f;  // /sqrt(96)
    const bool m = (ax != 0.0f) && (dq > th);
    s[kk] = m ? (0.5f * dq + 0.5f * ax) : dq;
  }
  float mx = s[0];
  for (int kk = 1; kk < EE; ++kk) mx = fmaxf(mx, s[kk]);
  float sum = 0.0f;
  for (int kk = 0; kk < EE; ++kk) {
    s[kk] = __expf(s[kk] - mx);
    sum += s[kk];
  }
  const float inv = 1.0f / sum;
  for (int c = 0; c < DD; ++c) {
    float acc = 0.0f;
    for (int kk = 0; kk < EE; ++kk)
      acc += s[kk] * v[((size_t)b * EE + kk) * DD + c];
    vhe[((size_t)b * EE + t) * DD + c] = acc * inv;
  }
}

// final concat([obs, t_he, v_he], axis=1) -> (B, 1224, 128)
__global__ void out_k(const float* __restrict__ obs, const float* __restrict__ the,
                      const float* __restrict__ vhe, float* __restrict__ out) {
  const size_t total = (size_t)BB * (NN + LL + EE) * DD;
  for (size_t t = (size_t)blockIdx.x * blockDim.x + threadIdx.x; t < total;
       t += (size_t)gridDim.x * blockDim.x) {
    const int c = (int)(t % DD);
    const size_t br = t / DD;
    const int row = (int)(br % (NN + LL + EE));
    const int b = (int)(br / (NN + LL + EE));
    float v;
    if (row < NN)
      v = obs[((size_t)b * NN + row) * DD + c];
    else if (row < NN + LL)
      v = the[((size_t)b * LL + (row - NN)) * DD + c];
    else
      v = vhe[((size_t)b * EE + (row - NN - LL)) * DD + c];
    out[t] = v;
  }
}

// =====================================================================
// Host side
// =====================================================================
struct MhabP { const float *Wq, *bq, *Wk, *bk, *Wv, *bv, *Wo, *bo; };
struct LayerP {
  MhabP n2t, n2v, slf;
  const float *qr, *qi, *qj, *qk, *qb;                 // quaternion h2n
  const float *spW, *spB, *spThr, *spLtau;             // spike
};
struct NetP {
  const float *irrWq, *irrbq, *irrWk, *irrbk, *irrWv, *irrbv, *irrThr;
  LayerP L[2];
};

static inline int ew_grid(size_t total) {
  size_t g = (total + 255) / 256;
  return (int)(g > 16384 ? 16384 : g);
}

static void gemm(int mode, const float* A, const float* W, const float* b,
                 float* out, int M, int K, const float* resid,
                 const float* rmask, const float* dtv, const float* ltau,
                 const float* thr, hipStream_t s) {
  dim3 g(M / 16, DD / 16);
  dim3 blk(32);
  switch (mode) {
    case 0: gemm_k<0><<<g, blk, 0, s>>>(A, W, b, out, M, K, resid, rmask, dtv, ltau, thr); break;
    case 1: gemm_k<1><<<g, blk, 0, s>>>(A, W, b, out, M, K, resid, rmask, dtv, ltau, thr); break;
    case 2: gemm_k<2><<<g, blk, 0, s>>>(A, W, b, out, M, K, resid, rmask, dtv, ltau, thr); break;
    default: gemm_k<3><<<g, blk, 0, s>>>(A, W, b, out, M, K, resid, rmask, dtv, ltau, thr); break;
  }
}

extern "C" void kernel_launch(void* const* d_in, const int* in_sizes, int n_in,
                              void* d_out, int out_size, void* d_ws,
                              size_t ws_size, hipStream_t stream) {
  const float* obs_in = (const float*)d_in[0];
  const float* the_in = (const float*)d_in[1];
  const float* vhe_in = (const float*)d_in[2];
  const int* tix = (const int*)d_in[3];
  const int* vix = (const int*)d_in[4];
  const float* tinc = (const float*)d_in[5];
  const float* vinc = (const float*)d_in[6];
  const float* xymf = (const float*)d_in[7];
  const float* xym = (const float*)d_in[8];
  const float* dtv = (const float*)d_in[10];
  auto F = [&](int i) { return (const float*)d_in[i]; };

  // ---- param pytree order: detect sorted-key (jax) vs insertion order ----
  NetP P;
  const bool sortedOrder = (n_in > 17 && in_sizes[17] == 1);
  if (sortedOrder) {
    P.irrWk = F(11); P.irrWq = F(12); P.irrWv = F(13);
    P.irrbk = F(14); P.irrbq = F(15); P.irrbv = F(16); P.irrThr = F(17);
    for (int i = 0; i < 2; ++i) {
      const int bs = 18 + 34 * i;
      LayerP& L = P.L[i];
      L.qb = F(bs + 0); L.qi = F(bs + 1); L.qj = F(bs + 2); L.qk = F(bs + 3); L.qr = F(bs + 4);
      MhabP* ms[3] = {&L.n2t, &L.n2v, &L.slf};
      for (int m = 0; m < 3; ++m) {
        const int o = bs + 5 + 8 * m;
        ms[m]->Wk = F(o + 0); ms[m]->Wo = F(o + 1); ms[m]->Wq = F(o + 2); ms[m]->Wv = F(o + 3);
        ms[m]->bk = F(o + 4); ms[m]->bo = F(o + 5); ms[m]->bq = F(o + 6); ms[m]->bv = F(o + 7);
      }
      L.spW = F(bs + 29); L.spB = F(bs + 30);
      L.spLtau = F(bs + 32); L.spThr = F(bs + 33);
    }
  } else {
    for (int i = 0; i < 2; ++i) {
      const int bs = 11 + 34 * i;
      LayerP& L = P.L[i];
      MhabP* ms[3] = {&L.n2t, &L.n2v, &L.slf};
      for (int m = 0; m < 3; ++m) {
        const int o = bs + 8 * m;
        ms[m]->Wq = F(o + 0); ms[m]->bq = F(o + 1); ms[m]->Wk = F(o + 2); ms[m]->bk = F(o + 3);
        ms[m]->Wv = F(o + 4); ms[m]->bv = F(o + 5); ms[m]->Wo = F(o + 6); ms[m]->bo = F(o + 7);
      }
      L.qr = F(bs + 24); L.qi = F(bs + 25); L.qj = F(bs + 26); L.qk = F(bs + 27); L.qb = F(bs + 28);
      L.spW = F(bs + 29); L.spB = F(bs + 30);
      L.spThr = F(bs + 31); L.spLtau = F(bs + 33);
    }
    P.irrWq = F(79); P.irrbq = F(80); P.irrWk = F(81); P.irrbk = F(82);
    P.irrWv = F(83); P.irrbv = F(84); P.irrThr = F(85);
  }

  // ---- workspace layout (floats) ----
  float* w = (float*)d_ws;
  const size_t SZ_OBS = (size_t)BB * NN * DD;
  const size_t SZ_THE = (size_t)BB * LL * DD;
  const size_t SZ_VHE = (size_t)BB * EE * DD;
  float* obs  = w;                        // 0
  float* the  = obs + SZ_OBS;
  float* vhe  = the + SZ_THE;
  float* tnew = vhe + SZ_VHE;
  float* vnew = tnew + SZ_THE;
  float* qbuf = vnew + SZ_VHE;
  float* kbuf = qbuf + SZ_OBS;
  float* vbuf = kbuf + SZ_OBS;
  float* obuf = vbuf + SZ_OBS;
  float* buf2 = obuf + SZ_OBS;
  float* kin  = buf2 + SZ_OBS;            // B*N*384 shared Kin / Xquat
  float* wqt  = kin + (size_t)BB * NN * 384;
  float* fine = wqt + (size_t)384 * DD;
  float* irq  = fine + (size_t)BB * EE * 96;
  float* irk  = irq + SZ_VHE;
  float* irv  = irk + SZ_VHE;
  (void)ws_size; (void)out_size;

  // ---- working copies of the recurrent state ----
  hipMemcpyAsync(obs, obs_in, SZ_OBS * sizeof(float), hipMemcpyDeviceToDevice, stream);
  hipMemcpyAsync(the, the_in, SZ_THE * sizeof(float), hipMemcpyDeviceToDevice, stream);
  hipMemcpyAsync(vhe, vhe_in, SZ_VHE * sizeof(float), hipMemcpyDeviceToDevice, stream);

  const int MN = BB * NN;   // 9216
  const int ML = BB * LL;   // 384
  const int ME = BB * EE;   // 192

  for (int layer = 0; layer < 2; ++layer) {
    const LayerP& L = P.L[layer];

    // ---------- n2t: t_upd = MHAB(t_he, [vg0 | obs], causal_t_inc) ----------
    build_kin2_k<<<ew_grid((size_t)MN * 256), 256, 0, stream>>>(vhe, EE, vix, obs, kin);
    gemm(0, the, L.n2t.Wq, L.n2t.bq, qbuf, ML, DD, nullptr, nullptr, nullptr, nullptr, nullptr, stream);
    gemm(0, kin, L.n2t.Wk, L.n2t.bk, kbuf, MN, 2 * DD, nullptr, nullptr, nullptr, nullptr, nullptr, stream);
    gemm(0, kin, L.n2t.Wv, L.n2t.bv, vbuf, MN, 2 * DD, nullptr, nullptr, nullptr, nullptr, nullptr, stream);
    attn_k<1><<<dim3(LL / 16, HH, BB), 32, 0, stream>>>(qbuf, kbuf, vbuf, obuf, LL, tinc, tix);
    gemm(1, obuf, L.n2t.Wo, L.n2t.bo, tnew, ML, DD, obuf, nullptr, nullptr, nullptr, nullptr, stream);

    // ---------- n2v: v_upd = MHAB(v_he, [tg0 | obs], v_inc) ----------
    build_kin2_k<<<ew_grid((size_t)MN * 256), 256, 0, stream>>>(the, LL, tix, obs, kin);
    gemm(0, vhe, L.n2v.Wq, L.n2v.bq, qbuf, ME, DD, nullptr, nullptr, nullptr, nullptr, nullptr, stream);
    gemm(0, kin, L.n2v.Wk, L.n2v.bk, kbuf, MN, 2 * DD, nullptr, nullptr, nullptr, nullptr, nullptr, stream);
    gemm(0, kin, L.n2v.Wv, L.n2v.bv, vbuf, MN, 2 * DD, nullptr, nullptr, nullptr, nullptr, nullptr, stream);
    attn_k<2><<<dim3((EE + 15) / 16, HH, BB), 32, 0, stream>>>(qbuf, kbuf, vbuf, obuf, EE, vinc, tix);
    gemm(1, obuf, L.n2v.Wo, L.n2v.bo, vnew, ME, DD, obuf, nullptr, nullptr, nullptr, nullptr, stream);

    hipMemcpyAsync(the, tnew, SZ_THE * sizeof(float), hipMemcpyDeviceToDevice, stream);
    hipMemcpyAsync(vhe, vnew, SZ_VHE * sizeof(float), hipMemcpyDeviceToDevice, stream);

    // ---------- self_upd: obs_upd = MHAB(obs, [tg|vg|obs], node_mask) ----------
    build_kin3_k<<<ew_grid((size_t)MN * 384), 256, 0, stream>>>(the, tix, vhe, vix, obs, kin);
    gemm(0, obs, L.slf.Wq, L.slf.bq, qbuf, MN, DD, nullptr, nullptr, nullptr, nullptr, nullptr, stream);
    gemm(0, kin, L.slf.Wk, L.slf.bk, kbuf, MN, 3 * DD, nullptr, nullptr, nullptr, nullptr, nullptr, stream);
    gemm(0, kin, L.slf.Wv, L.slf.bv, vbuf, MN, 3 * DD, nullptr, nullptr, nullptr, nullptr, nullptr, stream);
    attn_k<0><<<dim3(NN / 16, HH, BB), 32, 0, stream>>>(qbuf, kbuf, vbuf, obuf, NN, xymf, tix);
    gemm(1, obuf, L.slf.Wo, L.slf.bo, buf2, MN, DD, obuf, nullptr, nullptr, nullptr, nullptr, stream);

    // ---------- obs = relu(obs + quat([tg|vg|obs_upd])) * md ----------
    build_quatw_k<<<ew_grid(384 * DD), 256, 0, stream>>>(L.qr, L.qi, L.qj, L.qk, wqt);
    build_kin3_k<<<ew_grid((size_t)MN * 384), 256, 0, stream>>>(the, tix, vhe, vix, buf2, kin);
    gemm(2, kin, wqt, L.qb, obs, MN, 3 * DD, obs, xymf, nullptr, nullptr, nullptr, stream);

    // ---------- spike gate: obs = obs*(1+spike)*md ----------
    gemm(3, obs, L.spW, L.spB, obuf, MN, DD, obs, xymf, dtv, L.spLtau, L.spThr, stream);
    hipMemcpyAsync(obs, obuf, SZ_OBS * sizeof(float), hipMemcpyDeviceToDevice, stream);

    // ---------- last layer: v_he = _irr(v_he, fine, fine) ----------
    if (layer == 1) {
      fine_k<<<ew_grid((size_t)BB * EE * 96), 256, 0, stream>>>(obs, xym, fine);
      gemm(0, vhe, P.irrWq, P.irrbq, irq, ME, DD, nullptr, nullptr, nullptr, nullptr, nullptr, stream);
      gemm(0, vhe, P.irrWk, P.irrbk, irk, ME, DD, nullptr, nullptr, nullptr, nullptr, nullptr, stream);
      gemm(0, vhe, P.irrWv, P.irrbv, irv, ME, DD, nullptr, nullptr, nullptr, nullptr, nullptr, stream);
      irr_k<<<dim3(BB), 32, 0, stream>>>(irq, irk, irv, fine, P.irrThr, vhe);
    }
  }

  // ---------- concat([obs, t_he, v_he], axis=1) ----------
  out_k<<<ew_grid((size_t)BB * (NN + LL + EE) * DD), 256, 0, stream>>>(
      obs, the, vhe, (float*)d_out);
}